// ConvUpDown_44555990729381
// MI455X (gfx1250) — compile-verified
//
#include <hip/hip_runtime.h>
#include <hip/hip_bf16.h>
#include <stddef.h>

typedef __attribute__((ext_vector_type(2))) float v2f;
typedef __attribute__((ext_vector_type(8))) float v8f;

// Input:  images [4,128,512,512] f32 NCHW, w [1,1,128,256] f32 (ci-major, co contiguous)
// Output: out    [4,256,256,256] f32 NCHW
//
// Per workgroup: one (n, oy, ox-tile of 32), all 256 co.
//   Stage 1: FIR(4x4, pad1) + 2x downsample -> LDS  X[ci=128][m=32]
//   Stage 2: WMMA f32 16x16x4 GEMM: D[co, px] = sum_ci W[ci][co] * X[ci][px]
__global__ __launch_bounds__(256)
void ConvUpDown_fused_kernel(const float* __restrict__ img,
                             const float* __restrict__ w,
                             float* __restrict__ out)
{
    __shared__ float Xs[128 * 32];   // 16 KB: X[ci][m]

    const int t = threadIdx.x;
    int b = blockIdx.x;
    const int oxt = b & 7;   b >>= 3;     // 8 tiles of 32 ox
    const int oy  = b & 255; b >>= 8;     // 256 rows
    const int n   = b;                    // 4 batches
    const int ox0 = oxt << 5;

    // ---------------- Stage 1: FIR + downsample into LDS ----------------
    // k2d[dy][dx] = k1[dy]*k1[dx]/64, k1=[1,3,3,1]; symmetric => flip is identity.
    const float kw[4] = {1.f / 8.f, 3.f / 8.f, 3.f / 8.f, 1.f / 8.f};
    const int iy0 = 2 * oy - 1;

    #pragma unroll
    for (int j = 0; j < 16; ++j) {
        const int e  = t + 256 * j;       // 4096 elements total
        const int ci = e >> 5;
        const int m  = e & 31;
        const int ix0 = 2 * (ox0 + m) - 1;
        const float* base = img + ((size_t)(n * 128 + ci) << 18); // *512*512
        float acc = 0.f;
        #pragma unroll
        for (int dy = 0; dy < 4; ++dy) {
            const int y = iy0 + dy;
            if ((unsigned)y < 512u) {
                const float* row = base + ((size_t)y << 9);
                float r = 0.f;
                #pragma unroll
                for (int dx = 0; dx < 4; ++dx) {
                    const int x = ix0 + dx;
                    if ((unsigned)x < 512u) r += kw[dx] * row[x];
                }
                acc += kw[dy] * r;
            }
        }
        Xs[(ci << 5) + m] = acc;
    }
    __syncthreads();

    // ---------------- Stage 2: WMMA GEMM ----------------
    const int lane = t & 31;
    const int wv   = t >> 5;          // 8 waves
    const int pi   = wv & 1;          // pixel subtile (16 px each)
    const int cg   = wv >> 1;         // co group: 64 co each
    const int mrow = lane & 15;       // row within 16 (co for A/D, px for B/D-N)
    const int hi   = lane >> 4;       // half-wave select
    const int kofs = hi << 1;         // A/B: lanes 16-31 carry K+2
    const int px   = (pi << 4) + mrow;

    v8f acc[4] = {v8f{}, v8f{}, v8f{}, v8f{}};

    for (int kb = 0; kb < 32; ++kb) {
        const int ci = (kb << 2) + kofs;

        // B fragment (X from LDS): VGPR v, lane -> K = 4kb + v + 2*hi, N = px
        v2f bx;
        bx.x = Xs[(ci << 5) + px];
        bx.y = Xs[((ci + 1) << 5) + px];

        // A fragments (W from global, L2-resident): lane -> M = co row, K = v + 2*hi
        const float* wp = w + (size_t)ci * 256 + (cg << 6) + mrow;
        v2f a0, a1, a2, a3;
        a0.x = wp[0];   a0.y = wp[256 + 0];
        a1.x = wp[16];  a1.y = wp[256 + 16];
        a2.x = wp[32];  a2.y = wp[256 + 32];
        a3.x = wp[48];  a3.y = wp[256 + 48];

        acc[0] = __builtin_amdgcn_wmma_f32_16x16x4_f32(false, a0, false, bx,
                                                       (short)0, acc[0], false, false);
        acc[1] = __builtin_amdgcn_wmma_f32_16x16x4_f32(false, a1, false, bx,
                                                       (short)0, acc[1], false, false);
        acc[2] = __builtin_amdgcn_wmma_f32_16x16x4_f32(false, a2, false, bx,
                                                       (short)0, acc[2], false, false);
        acc[3] = __builtin_amdgcn_wmma_f32_16x16x4_f32(false, a3, false, bx,
                                                       (short)0, acc[3], false, false);
    }

    // ---------------- Store: D VGPR j -> co = base + j (+8 for hi half), N lane -> ox
    const int ox = ox0 + px;
    float* outb = out + ((size_t)n << 24) + ((size_t)oy << 8) + ox;
    #pragma unroll
    for (int s = 0; s < 4; ++s) {
        const int cobase = (cg << 6) + (s << 4) + (hi << 3);
        #pragma unroll
        for (int jj = 0; jj < 8; ++jj) {
            outb[(size_t)(cobase + jj) << 16] = acc[s][jj];
        }
    }
}

extern "C" void kernel_launch(void* const* d_in, const int* in_sizes, int n_in,
                              void* d_out, int out_size, void* d_ws, size_t ws_size,
                              hipStream_t stream)
{
    (void)in_sizes; (void)n_in; (void)d_ws; (void)ws_size; (void)out_size;
    const float* img = (const float*)d_in[0];   // 4*128*512*512
    const float* w   = (const float*)d_in[1];   // 128*256
    float* out       = (float*)d_out;           // 4*256*256*256

    dim3 grid(8 * 256 * 4);   // ox-tiles * oy * n = 8192
    dim3 block(256);
    hipLaunchKernelGGL(ConvUpDown_fused_kernel, grid, block, 0, stream, img, w, out);
}